// GraphDenoiser_44573170598305
// MI455X (gfx1250) — compile-verified
//
#include <hip/hip_runtime.h>
#include <hip/hip_bf16.h>

typedef __attribute__((ext_vector_type(2))) float        v2f;
typedef __attribute__((ext_vector_type(8))) float        v8f;
typedef __attribute__((ext_vector_type(4))) unsigned int v4u;
typedef __attribute__((ext_vector_type(4))) int          v4i;
typedef __attribute__((ext_vector_type(8))) int          v8i;

#define BATCH 4
#define NNODE 512
#define DIM   128

// ---------------------------------------------------------------------------
// Kernel A: precompute v[d] = (e1-e0)^T W1a  and  c0[b][d] = e0^T W1a +
// nc^T W1b + t_emb[b]^T W1c + b1.  Collapses the whole 384x128 GEMM to rank-1.
// ---------------------------------------------------------------------------
__global__ __launch_bounds__(128) void prep_kernel(
    const int* __restrict__ t, const float* __restrict__ time_embed,
    const float* __restrict__ no_cond, const float* __restrict__ edge_embed,
    const float* __restrict__ W1, const float* __restrict__ b1,
    float* __restrict__ vout, float* __restrict__ c0out)
{
    int d = threadIdx.x;   // 0..127, coalesced across W1 rows
    float acc_v = 0.f, acc_s0 = 0.f, acc_nc = 0.f;
    for (int k = 0; k < DIM; ++k) {
        float w = W1[k * DIM + d];
        acc_v  += (edge_embed[DIM + k] - edge_embed[k]) * w;
        acc_s0 += edge_embed[k] * w;
        acc_nc += no_cond[k] * W1[(DIM + k) * DIM + d];
    }
    vout[d] = acc_v;
    float base = acc_s0 + acc_nc + b1[d];
    for (int bb = 0; bb < BATCH; ++bb) {
        const float* te = time_embed + (size_t)t[bb] * DIM;
        float acc_t = 0.f;
        for (int k = 0; k < DIM; ++k)
            acc_t += te[k] * W1[(2 * DIM + k) * DIM + d];
        c0out[bb * DIM + d] = base + acc_t;
    }
}

// ---------------------------------------------------------------------------
// Kernel B: 4 waves (128 threads) per 16-row tile.
//   0) TDM: DMA the 16x512 int32 adj tile into LDS (tensor_load_to_lds,
//      D# per CDNA5 ISA; wave 0 issues, s_wait_tensorcnt 0)
//   1) p[m] = mean_j adj[row,j]               (LDS int4 + shfl reduction)
//   2) h[m][k] = relu(c0[b][k] + p[m]*v[k])   -> LDS (padded stride)
//   3) H_fused = relu(h @ W2 + b2) via V_WMMA_F32_16X16X4_F32, K=128,
//      waves split the 8 column tiles (2 each)
//   4) li/lj = H_fused @ Wc halves            -> workspace
// ---------------------------------------------------------------------------
#define HSTRIDE 132   // 128 + 4 pad: breaks LDS bank alignment across rows

__global__ __launch_bounds__(128) void fused_kernel(
    const int* __restrict__ adj, const float* __restrict__ c0,
    const float* __restrict__ v,  const float* __restrict__ W2,
    const float* __restrict__ b2, const float* __restrict__ Wc,
    float* __restrict__ li, float* __restrict__ lj)
{
    __shared__ int   sh_adj[16 * NNODE];     // 32 KB TDM destination
    __shared__ float sh_h [16 * HSTRIDE];
    __shared__ float sh_hf[16 * HSTRIDE];
    __shared__ float sh_p [16];

    const int tid  = threadIdx.x;       // 0..127
    const int lane = tid & 31;
    const int wave = tid >> 5;
    const int tile = blockIdx.x;        // 0..127
    const int r0   = tile * 16;         // first global row (b*N + n)
    const int b    = r0 >> 9;           // tile never straddles a batch (512%16==0)

    // ---- 0) DMA adj tile -> LDS via Tensor Data Mover ----
#if __has_builtin(__builtin_amdgcn_tensor_load_to_lds)
    if (tid < 32) {                     // TDM ignores EXEC: issue from wave 0 only
        unsigned long long gaddr = (unsigned long long)(uintptr_t)(adj + (size_t)r0 * NNODE);
        unsigned int lds_off = (unsigned int)(uintptr_t)(void*)sh_adj;  // LDS aperture: low 32 bits
        // D# group 0: count=1 | lds_addr | global_addr[56:0] | type=2
        v4u g0;
        g0.x = 1u;                                  // count=1, is_restore=0, gather off
        g0.y = lds_off;                             // lds_addr (bytes)
        g0.z = (unsigned int)(gaddr & 0xFFFFFFFFu); // global_addr[31:0]
        g0.w = (unsigned int)((gaddr >> 32) & 0x1FFFFFFu) | (2u << 30); // addr[56:32], type=2
        // D# group 1: data_size=4B; tensor 512x2048; tile 512x16; stride0=512
        v8i g1;
        g1[0] = (int)(2u << 16);                    // data_size=2 (4B); mask=0 (not in cluster)
        g1[1] = (int)(512u << 16);                  // tensor_dim0[15:0]=512
        g1[2] = (int)(2048u << 16);                 // tensor_dim0[31:16]=0 | tensor_dim1[15:0]=2048
        g1[3] = (int)(512u << 16);                  // tensor_dim1[31:16]=0 | tile_dim0=512
        g1[4] = 16;                                 // tile_dim1=16, tile_dim2=0 (2D tile)
        g1[5] = 512;                                // tensor_dim0_stride[31:0]=512
        g1[6] = 0;                                  // stride0[47:32]=0, stride1 lo=0
        g1[7] = 0;
        v4i z4 = {0, 0, 0, 0};
#if defined(__clang_major__) && (__clang_major__ >= 23)
        v8i z8 = {0, 0, 0, 0, 0, 0, 0, 0};
        __builtin_amdgcn_tensor_load_to_lds(g0, g1, z4, z4, z8, 0);
#else
        __builtin_amdgcn_tensor_load_to_lds(g0, g1, z4, z4, 0);
#endif
        __builtin_amdgcn_s_wait_tensorcnt(0);
    }
#else
    // Fallback: cooperative copy
    {
        const int4* src = (const int4*)(adj + (size_t)r0 * NNODE);
        int4* dst = (int4*)sh_adj;
        for (int idx = tid; idx < 16 * NNODE / 4; idx += 128) dst[idx] = src[idx];
    }
#endif
    __syncthreads();

    // ---- 1) p for the 16 rows: wave w reduces rows w, w+4, w+8, w+12 ----
    const float inv_n = 1.0f / (float)NNODE;
    for (int m = wave; m < 16; m += 4) {
        const int4* r4 = (const int4*)(sh_adj + m * NNODE);
        int s = 0;
        #pragma unroll
        for (int i = 0; i < 4; ++i) {
            int4 val = r4[lane + 32 * i];
            s += val.x + val.y + val.z + val.w;
        }
        #pragma unroll
        for (int off = 16; off > 0; off >>= 1) s += __shfl_xor(s, off, 32);
        if (lane == 0) sh_p[m] = (float)s * inv_n;
    }
    __syncthreads();

    // ---- 2) h tile: relu(c0 + p*v), rank-1 replacement of GEMM1 ----
    const float* c0b = c0 + b * DIM;
    for (int idx = tid; idx < 16 * DIM; idx += 128) {
        int m = idx >> 7, k = idx & (DIM - 1);
        float val = c0b[k] + sh_p[m] * v[k];
        sh_h[m * HSTRIDE + k] = val > 0.f ? val : 0.f;
    }
    __syncthreads();

    // ---- 3) GEMM2 via fp32 WMMA (16x16x4), K = 128; wave w -> ct = w, w+4 ----
    // A (16x4 f32): lanes 0-15 -> M=lane, K={0,1}; lanes 16-31 -> M=lane-16, K={2,3}
    // B (4x16 f32): VGPR0 = K=0 (lanes 0-15) / K=2 (lanes 16-31); VGPR1 = K=1 / K=3
    // D (16x16):    VGPR j -> M=j (lanes 0-15), M=8+j (lanes 16-31); N = lane&15
    const int mlane = lane & 15;
    const int koff  = (lane >> 4) * 2;
    const int mbase = (lane >> 4) * 8;

    for (int ct = wave; ct < 8; ct += 4) {
        const int d = ct * 16 + mlane;
        v8f acc = {0.f, 0.f, 0.f, 0.f, 0.f, 0.f, 0.f, 0.f};
        #pragma unroll 4
        for (int ko = 0; ko < 32; ++ko) {
            const int k = ko * 4 + koff;
            v2f a;
            a.x = sh_h[mlane * HSTRIDE + k];
            a.y = sh_h[mlane * HSTRIDE + k + 1];
            v2f bfrag;
            bfrag.x = W2[k * DIM + d];
            bfrag.y = W2[(k + 1) * DIM + d];
            acc = __builtin_amdgcn_wmma_f32_16x16x4_f32(
                false, a, false, bfrag, (short)0, acc, false, false);
        }
        const float bias = b2[d];
        #pragma unroll
        for (int j = 0; j < 8; ++j) {
            float val = acc[j] + bias;
            sh_hf[(mbase + j) * HSTRIDE + d] = val > 0.f ? val : 0.f;
        }
    }
    __syncthreads();

    // ---- 4) classifier halves: threads 0-15 -> li, 16-31 -> lj ----
    if (tid < 32) {
        const int m = tid & 15;
        const float* hr = sh_hf + m * HSTRIDE;
        const float* wc = (tid < 16) ? Wc : (Wc + 2 * DIM);  // Wc[256,2] row-major
        float a0 = 0.f, a1 = 0.f;
        for (int dd = 0; dd < DIM; ++dd) {
            float hv = hr[dd];
            a0 += hv * wc[dd * 2 + 0];
            a1 += hv * wc[dd * 2 + 1];
        }
        float* outp = (tid < 16) ? li : lj;
        const int row = r0 + m;
        outp[row * 2 + 0] = a0;
        outp[row * 2 + 1] = a1;
    }
}

// ---------------------------------------------------------------------------
// Kernel C: logits[b,i,j,:] = li[b,i,:] + lj[b,j,:] + bc  (8 MB stream, the
// bandwidth-dominant phase; coalesced nontemporal float2 stores)
// ---------------------------------------------------------------------------
__global__ __launch_bounds__(256) void expand_kernel(
    const float* __restrict__ li, const float* __restrict__ lj,
    const float* __restrict__ bc, float* __restrict__ out)
{
    const int row = blockIdx.x;              // b*N + i, 0..2047
    const float li0 = li[row * 2 + 0] + bc[0];
    const float li1 = li[row * 2 + 1] + bc[1];
    const int b = row >> 9;
    const v2f* lj2 = (const v2f*)lj + (size_t)b * NNODE;
    v2f* orow = (v2f*)out + (size_t)row * NNODE;
    for (int j = threadIdx.x; j < NNODE; j += 256) {
        v2f l = lj2[j];
        v2f o;
        o.x = li0 + l.x;
        o.y = li1 + l.y;
        __builtin_nontemporal_store(o, &orow[j]);   // write-once stream: NT hint
    }
}

// ---------------------------------------------------------------------------
extern "C" void kernel_launch(void* const* d_in, const int* in_sizes, int n_in,
                              void* d_out, int out_size, void* d_ws, size_t ws_size,
                              hipStream_t stream)
{
    (void)in_sizes; (void)n_in; (void)out_size; (void)ws_size;
    const int*   adj        = (const int*)  d_in[0];
    const int*   t          = (const int*)  d_in[1];
    const float* time_embed = (const float*)d_in[2];
    const float* no_cond    = (const float*)d_in[3];
    const float* edge_embed = (const float*)d_in[4];
    const float* W1         = (const float*)d_in[5];
    const float* b1         = (const float*)d_in[6];
    const float* W2         = (const float*)d_in[7];
    const float* b2         = (const float*)d_in[8];
    const float* Wc         = (const float*)d_in[9];
    const float* bc         = (const float*)d_in[10];

    // workspace layout (floats): v[128] | c0[4*128] | li[2048*2] | lj[2048*2]
    float* w   = (float*)d_ws;
    float* v   = w;
    float* c0  = w + 128;
    float* li  = w + 128 + BATCH * DIM;                  // +640
    float* lj  = li + BATCH * NNODE * 2;                 // +4096 floats

    prep_kernel  <<<1, 128, 0, stream>>>(t, time_embed, no_cond, edge_embed, W1, b1, v, c0);
    fused_kernel <<<(BATCH * NNODE) / 16, 128, 0, stream>>>(adj, c0, v, W2, b2, Wc, li, lj);
    expand_kernel<<<BATCH * NNODE, 256, 0, stream>>>(li, lj, bc, (float*)d_out);
}